// Unpool_75857712382623
// MI455X (gfx1250) — compile-verified
//
#include <hip/hip_runtime.h>

// ---------------------------------------------------------------------------
// Icosphere unpool (mean over 2 parents):  out[r, :] = 0.5*(F[p0[r],:] + F[p1[r],:])
//
// Pure gather + average: memory-bound. HBM traffic floor on MI455X:
//   168 MB out-writes + ~42 MB feature reads (table fits in 192MB L2) + 5 MB idx
//   => ~9.2 us at 23.3 TB/s. No matmul -> WMMA cannot help; the CDNA5 feature
// that matters is the async-to-LDS gather engine (ASYNCcnt) + cache hints:
//   - per-lane global_load_async_to_lds_b128 gathers have no VGPR destination,
//     so a wave keeps both parent-row fetches in flight with zero reg hazards
//   - non-temporal stores keep the 168 MB output stream from evicting the
//     L2-resident feature table; indices are streamed NT as well.
// ---------------------------------------------------------------------------

typedef __attribute__((ext_vector_type(4))) float v4f;
typedef __attribute__((ext_vector_type(4))) int   v4i;   // builtin's pointee type
typedef __attribute__((ext_vector_type(2))) int   v2i;

#define CCH            64          // channels
#define ROWS_PER_BLOCK 16          // 16 lanes per row * 16 rows = 256 threads
#define THREADS        256

#if defined(__has_builtin)
#  if __has_builtin(__builtin_amdgcn_global_load_async_to_lds_b128) && \
      __has_builtin(__builtin_amdgcn_s_wait_asynccnt)
#    define USE_ASYNC_GATHER 1
#  else
#    define USE_ASYNC_GATHER 0
#  endif
#else
#  define USE_ASYNC_GATHER 0
#endif

__global__ __launch_bounds__(THREADS) void unpool_mean_kernel(
    const float* __restrict__ feat,   // [N_COARSE, 64]
    const int*   __restrict__ pidx,   // [N_FINE, 2]
    float*       __restrict__ out,    // [N_FINE, 64]
    int n_fine)
{
    const int tid = threadIdx.x;
    const int row = blockIdx.x * ROWS_PER_BLOCK + (tid >> 4);  // fine vertex
    const int ch4 = tid & 15;                                  // float4 chunk 0..15

#if USE_ASYNC_GATHER
    __shared__ v4f stage[THREADS * 2];   // 8 KB: 2 private 16B slots per thread
#endif

    if (row >= n_fine) return;

    // Both parent indices in one 8B NT load (streamed once, don't cache-pollute).
    const v2i p = __builtin_nontemporal_load((const v2i*)pidx + row);

    const v4f* __restrict__ f0 = (const v4f*)(feat + (size_t)p.x * CCH) + ch4;
    const v4f* __restrict__ f1 = (const v4f*)(feat + (size_t)p.y * CCH) + ch4;

    v4f a, b;
#if USE_ASYNC_GATHER
    // Fire-and-forget 128b gathers into this thread's private LDS slots.
    v4f* s0 = &stage[tid * 2 + 0];
    v4f* s1 = &stage[tid * 2 + 1];
    __builtin_amdgcn_global_load_async_to_lds_b128((v4i*)f0, (v4i*)s0, 0, 0);
    __builtin_amdgcn_global_load_async_to_lds_b128((v4i*)f1, (v4i*)s1, 0, 0);
    __builtin_amdgcn_s_wait_asynccnt(0);
    __asm__ volatile("" ::: "memory");   // compiler barrier: LDS now valid
    a = *s0;                              // ds_load_b128
    b = *s1;
#else
    a = *f0;                              // global_load_b128
    b = *f1;
#endif

    v4f r = 0.5f * (a + b);

    // Output is written once and never re-read by this dispatch: stream it (NT)
    // so it does not evict the L2-resident feature table.
    __builtin_nontemporal_store(r, (v4f*)out + (size_t)row * (CCH / 4) + ch4);
}

extern "C" void kernel_launch(void* const* d_in, const int* in_sizes, int n_in,
                              void* d_out, int out_size, void* d_ws, size_t ws_size,
                              hipStream_t stream) {
    const float* feat = (const float*)d_in[0];   // float32 [N_COARSE*64]
    const int*   pidx = (const int*)d_in[1];     // integer  [N_FINE*2]
    float*       out  = (float*)d_out;           // float32 [N_FINE*64]

    const int n_fine = in_sizes[1] / 2;          // parent_idx has N_FINE*2 elems
    const int blocks = (n_fine + ROWS_PER_BLOCK - 1) / ROWS_PER_BLOCK;

    hipLaunchKernelGGL(unpool_mean_kernel, dim3(blocks), dim3(THREADS), 0, stream,
                       feat, pidx, out, n_fine);
}